// MaxUnpooling2D_24584392802909
// MI455X (gfx1250) — compile-verified
//
#include <hip/hip_runtime.h>

// MaxUnpooling2D scatter-add for MI455X (gfx1250, wave32).
//
// Roofline: ~335 MB of HBM traffic (134 MB zero-fill stores + 67 MB input
// reads + <=134 MB writeback) -> ~14 us floor at 23.3 TB/s. The 134 MB output
// fits in the 192 MB L2, so the zero-fill (regular-temporal b128 stores)
// leaves it L2-resident and the 8.4M random fp32 scatter atomics are resolved
// on-chip by the L2 atomic units. Duplicate indices must sum -> atomics are
// mandatory; we use the native no-return device-scope global_atomic_add_f32
// (STOREcnt-tracked, no VGPR writeback, max outstanding ops per wave).

__device__ __forceinline__ void global_atomic_add_f32_nr(float* addr, float val) {
    asm volatile("global_atomic_add_f32 %0, %1, off scope:SCOPE_DEV"
                 :
                 : "v"(addr), "v"(val)
                 : "memory");
}

// Zero-fill: block-tiled, 4 x global_store_b128 per thread, fully coalesced.
__global__ void zero_fill_b128x4(float4* __restrict__ out, long long n4) {
    const int B = blockDim.x;
    long long tile = (long long)blockIdx.x * (4LL * B) + threadIdx.x;
    const float4 z = make_float4(0.0f, 0.0f, 0.0f, 0.0f);
#pragma unroll
    for (int j = 0; j < 4; ++j) {
        long long i = tile + (long long)j * B;
        if (i < n4) out[i] = z;
    }
}

// Scatter: 8 elements/thread as two contiguous float4/int4 streams
// (offsets i and i+ngroups) -> both b128 loads perfectly coalesced,
// 8 outstanding no-return atomics per lane.
__global__ void scatter_add8(const float4* __restrict__ upd4,
                             const int4* __restrict__ idx4,
                             float* __restrict__ out,
                             long long ngroups,     // pairs of float4 groups
                             long long n_total) {
    long long i = (long long)blockIdx.x * blockDim.x + threadIdx.x;
    if (i < ngroups) {
        float4 u0 = upd4[i];              // global_load_b128 (stream 0)
        float4 u1 = upd4[i + ngroups];    // global_load_b128 (stream 1)
        int4   m0 = idx4[i];              // global_load_b128
        int4   m1 = idx4[i + ngroups];    // global_load_b128
        global_atomic_add_f32_nr(out + (long long)m0.x, u0.x);
        global_atomic_add_f32_nr(out + (long long)m0.y, u0.y);
        global_atomic_add_f32_nr(out + (long long)m0.z, u0.z);
        global_atomic_add_f32_nr(out + (long long)m0.w, u0.w);
        global_atomic_add_f32_nr(out + (long long)m1.x, u1.x);
        global_atomic_add_f32_nr(out + (long long)m1.y, u1.y);
        global_atomic_add_f32_nr(out + (long long)m1.z, u1.z);
        global_atomic_add_f32_nr(out + (long long)m1.w, u1.w);
    }
    // Scalar tail for n_total % 8 != 0 (empty at the reference shapes).
    if (i == 0) {
        const float* upd = (const float*)upd4;
        const int*   idx = (const int*)idx4;
        for (long long k = ngroups * 8; k < n_total; ++k) {
            global_atomic_add_f32_nr(out + (long long)idx[k], upd[k]);
        }
    }
}

extern "C" void kernel_launch(void* const* d_in, const int* in_sizes, int n_in,
                              void* d_out, int out_size, void* d_ws, size_t ws_size,
                              hipStream_t stream) {
    const float* updates = (const float*)d_in[0];
    const int*   mask    = (const int*)d_in[1];
    float*       out     = (float*)d_out;

    const long long n       = (long long)in_sizes[0];   // 8,388,608 updates
    const long long n4      = n / 4;                    // float4 groups
    const long long ngroups = n4 / 2;                   // pairs of groups (8 elems)
    const long long out_n   = (long long)out_size;      // 33,554,432 floats
    const long long out_n4  = out_n / 4;                // float4 groups

    const int block = 256;  // 8 wave32 waves

    // 1) Zero output every call (harness poisons once; replays accumulate).
    {
        long long per_block = 4LL * block;
        long long grid = (out_n4 + per_block - 1) / per_block;
        if (grid == 0) grid = 1;
        zero_fill_b128x4<<<(unsigned)grid, block, 0, stream>>>((float4*)out, out_n4);
    }

    // 2) Scatter-add (stream-ordered after fill; output L2-resident).
    {
        long long grid = (ngroups + block - 1) / block;
        if (grid == 0) grid = 1;  // tail-only case
        scatter_add8<<<(unsigned)grid, block, 0, stream>>>(
            (const float4*)updates, (const int4*)mask, out, ngroups, n);
    }
}